// BionetworkModel_27668179321232
// MI455X (gfx1250) — compile-verified
//
#include <hip/hip_runtime.h>
#include <hip/hip_bf16.h>
#include <stdint.h>

// Problem constants (match reference)
constexpr int kB      = 64;
constexpr int kNIn    = 128;
constexpr int kNOut   = 256;
constexpr int kNodes  = 20000;
constexpr int kEdges  = 320000;
constexpr int kIters  = 150;
constexpr float kLeak = 0.01f;

// ---------------------------------------------------------------------------
// CDNA5 async global->LDS copy helpers (gfx1250). Inline asm is portable
// across ROCm-7.2 clang-22 and upstream clang-23 (bypasses clang builtins).
// ---------------------------------------------------------------------------
__device__ __forceinline__ unsigned ldsOffset(const void* p) {
  // Generic pointers to __shared__ carry the LDS offset in the low 32 bits.
  return (unsigned)(unsigned long long)p;
}

__device__ __forceinline__ void asyncLoadB64(unsigned ldsAddr, const void* gaddr) {
  asm volatile("global_load_async_to_lds_b64 %0, %1, off"
               :: "v"(ldsAddr), "v"(gaddr) : "memory");
}

__device__ __forceinline__ void waitAsync0() {
  asm volatile("s_wait_asynccnt 0" ::: "memory");
}

// ---------------------------------------------------------------------------
// Small utility kernels
// ---------------------------------------------------------------------------
__global__ void k_zeroI32(int* p, int n) {
  int i = blockIdx.x * blockDim.x + threadIdx.x;
  if (i < n) p[i] = 0;
}

__global__ void k_zeroF32(float* p, int n) {
  int i = blockIdx.x * blockDim.x + threadIdx.x;
  if (i < n) p[i] = 0.0f;
}

__global__ void k_hist(const int* __restrict__ rows, int* counts) {
  int e = blockIdx.x * blockDim.x + threadIdx.x;
  if (e < kEdges) atomicAdd(&counts[rows[e]], 1);
}

// One 1024-thread block scans the 20000 row counts (chunked + Hillis-Steele).
// counts aliases cursor: each thread reads its own chunk before overwriting it.
__global__ __launch_bounds__(1024) void k_scan(int* counts /*==cursor*/,
                                               int* rowStart, int* cursor) {
  __shared__ int part[1024];
  const int CH = (kNodes + 1023) / 1024;  // 20
  int t  = threadIdx.x;
  int lo = t * CH;
  int hi = lo + CH; if (hi > kNodes) hi = kNodes; if (lo > kNodes) lo = kNodes;
  int s = 0;
  for (int i = lo; i < hi; ++i) s += counts[i];
  part[t] = s;
  __syncthreads();
  for (int off = 1; off < 1024; off <<= 1) {
    int v = (t >= off) ? part[t - off] : 0;
    __syncthreads();
    part[t] += v;
    __syncthreads();
  }
  int run = (t == 0) ? 0 : part[t - 1];
  for (int i = lo; i < hi; ++i) {
    int c = counts[i];           // read before overwrite (counts==cursor)
    rowStart[i] = run;
    cursor[i]   = run;
    run += c;
  }
  if (t == 1023) rowStart[kNodes] = kEdges;
}

__global__ void k_scatter(const int* __restrict__ rows, int* cursor, int* eIdx) {
  int e = blockIdx.x * blockDim.x + threadIdx.x;
  if (e < kEdges) {
    int p = atomicAdd(&cursor[rows[e]], 1);
    eIdx[p] = e;
  }
}

// Per-row insertion sort on original edge index -> canonical (deterministic)
// order, then gather packed (col, weight-bits) pairs into the CSR array.
__global__ void k_sortgather(const int* __restrict__ rowStart, int* eIdx,
                             const int* __restrict__ cols,
                             const float* __restrict__ recW,
                             int2* __restrict__ ePair) {
  int r = blockIdx.x * blockDim.x + threadIdx.x;
  if (r >= kNodes) return;
  int s = rowStart[r], e = rowStart[r + 1];
  for (int i = s + 1; i < e; ++i) {
    int key = eIdx[i];
    int j = i - 1;
    while (j >= s && eIdx[j] > key) { eIdx[j + 1] = eIdx[j]; --j; }
    eIdx[j + 1] = key;
  }
  for (int p = s; p < e; ++p) {
    int ei = eIdx[p];
    ePair[p] = make_int2(cols[ei], __float_as_int(recW[ei]));
  }
}

__global__ void k_binInit(const float* __restrict__ biases, float* __restrict__ bin) {
  int i = blockIdx.x * blockDim.x + threadIdx.x;
  if (i < kNodes * kB) bin[i] = biases[i >> 6];
}

// Scatter input projection; sequential over j => last-write-wins for dup idx.
__global__ void k_binScatter(const float* __restrict__ x,
                             const float* __restrict__ inW,
                             const int* __restrict__ inIdx,
                             const float* __restrict__ biases,
                             float* __restrict__ bin) {
  int b = threadIdx.x;  // 64 threads, 1 block
  for (int j = 0; j < kNIn; ++j) {
    int n = inIdx[j];
    bin[n * kB + b] = biases[n] + inW[j] * x[b * kNIn + j];
  }
}

// ---------------------------------------------------------------------------
// Fused SpMV + bias + MML activation. One wave per row; lane ln owns batch
// columns 2*ln and 2*ln+1 (float2). The row's (col,w) edge list is staged into
// LDS with one CDNA5 async b64 copy per edge per wave (instead of 32 redundant
// per-lane loads), synchronized with s_wait_asynccnt -- no barrier needed.
// Unroll-by-4 keeps several independent 256B h-gathers in flight per wave.
// ---------------------------------------------------------------------------
__device__ __forceinline__ float mmlAct(float v) {
  v = (v < 0.0f) ? v * kLeak : v;
  return (v > 0.5f) ? (1.0f - 0.25f / v) : v;
}

constexpr int kWavesPerBlock = 8;

__global__ __launch_bounds__(kWavesPerBlock * 32)
void k_iter(const float* __restrict__ hin, float* __restrict__ hout,
            const float* __restrict__ bin, const int* __restrict__ rowStart,
            const int2* __restrict__ ePair) {
  __shared__ int2 sPair[kWavesPerBlock][32];

  const int wv  = threadIdx.x >> 5;
  const int ln  = threadIdx.x & 31;
  const int row = blockIdx.x * kWavesPerBlock + wv;   // wave-uniform
  if (row >= kNodes) return;                          // whole wave exits

  const int s    = rowStart[row];
  const int eEnd = rowStart[row + 1];

  float2 acc = *(const float2*)(bin + row * kB + 2 * ln);

  const unsigned ldsPair = ldsOffset(&sPair[wv][ln]);

  for (int base = s; base < eEnd; base += 32) {
    int n = eEnd - base;
    if (n > 32) n = 32;
    if (ln < n) {
      asyncLoadB64(ldsPair, ePair + base + ln);
    }
    waitAsync0();  // same wave issued the copies; no barrier needed
#pragma unroll 4
    for (int e = 0; e < n; ++e) {
      const int2  pr = sPair[wv][e];                 // LDS b64 broadcast read
      const float w  = __int_as_float(pr.y);
      const float2* hp = (const float2*)(hin + (size_t)pr.x * kB);
      const float2 hv = hp[ln];                      // 256B coalesced gather
      acc.x = fmaf(w, hv.x, acc.x);
      acc.y = fmaf(w, hv.y, acc.y);
    }
  }
  float2 r;
  r.x = mmlAct(acc.x);
  r.y = mmlAct(acc.y);
  *(float2*)(hout + row * kB + 2 * ln) = r;
}

__global__ void k_out(const float* __restrict__ h, const float* __restrict__ outW,
                      const int* __restrict__ outIdx, float* __restrict__ out) {
  int i = blockIdx.x * blockDim.x + threadIdx.x;
  if (i >= kB * kNOut) return;
  int b = i / kNOut;
  int o = i % kNOut;
  out[i] = outW[o] * h[(size_t)outIdx[o] * kB + b];
}

// ---------------------------------------------------------------------------
// Host-side launch
// ---------------------------------------------------------------------------
static inline size_t alignUp(size_t v, size_t a) { return (v + a - 1) & ~(a - 1); }

extern "C" void kernel_launch(void* const* d_in, const int* in_sizes, int n_in,
                              void* d_out, int out_size, void* d_ws, size_t ws_size,
                              hipStream_t stream) {
  (void)in_sizes; (void)n_in; (void)out_size;

  const float* x      = (const float*)d_in[0];
  const float* inW    = (const float*)d_in[1];
  const float* outW   = (const float*)d_in[2];
  const float* recW   = (const float*)d_in[3];
  const float* biases = (const float*)d_in[4];
  const int*   rows   = (const int*)d_in[5];
  const int*   cols   = (const int*)d_in[6];
  const int*   inIdx  = (const int*)d_in[7];
  const int*   outIdx = (const int*)d_in[8];
  float* out = (float*)d_out;

  // Workspace layout (~19.4 MB total)
  char* ws = (char*)d_ws;
  size_t off = 0;
  int*   rowStart = (int*)(ws + off);   off = alignUp(off + (kNodes + 1) * 4, 256);
  int*   cursor   = (int*)(ws + off);   off = alignUp(off + kNodes * 4, 256);
  int*   eIdx     = (int*)(ws + off);   off = alignUp(off + kEdges * 4, 256);
  int2*  ePair    = (int2*)(ws + off);  off = alignUp(off + (size_t)kEdges * 8, 256);
  float* bin      = (float*)(ws + off); off = alignUp(off + (size_t)kNodes * kB * 4, 256);
  float* h0       = (float*)(ws + off); off = alignUp(off + (size_t)kNodes * kB * 4, 256);
  float* h1       = (float*)(ws + off); off = alignUp(off + (size_t)kNodes * kB * 4, 256);
  (void)ws_size;

  // --- Build deterministic CSR ---
  k_zeroI32<<<(kNodes + 255) / 256, 256, 0, stream>>>(cursor, kNodes);
  k_hist<<<(kEdges + 255) / 256, 256, 0, stream>>>(rows, cursor);
  k_scan<<<1, 1024, 0, stream>>>(cursor, rowStart, cursor);
  k_scatter<<<(kEdges + 255) / 256, 256, 0, stream>>>(rows, cursor, eIdx);
  k_sortgather<<<(kNodes + 255) / 256, 256, 0, stream>>>(rowStart, eIdx, cols, recW,
                                                         ePair);

  // --- b_in = bias + input scatter; h0 = 0 ---
  const int nbElems = kNodes * kB;
  k_binInit<<<(nbElems + 255) / 256, 256, 0, stream>>>(biases, bin);
  k_binScatter<<<1, kB, 0, stream>>>(x, inW, inIdx, biases, bin);
  k_zeroF32<<<(nbElems + 255) / 256, 256, 0, stream>>>(h0, nbElems);

  // --- 150 recurrent iterations, double-buffered ---
  float* hc = h0;
  float* hn = h1;
  const int iterBlocks = (kNodes + kWavesPerBlock - 1) / kWavesPerBlock;
  for (int it = 0; it < kIters; ++it) {
    k_iter<<<iterBlocks, kWavesPerBlock * 32, 0, stream>>>(hc, hn, bin, rowStart,
                                                           ePair);
    float* t = hc; hc = hn; hn = t;
  }

  // --- Output projection ---
  k_out<<<(kB * kNOut + 255) / 256, 256, 0, stream>>>(hc, outW, outIdx, out);
}